// SynthesisFNO_35347580846473
// MI455X (gfx1250) — compile-verified
//
#include <hip/hip_runtime.h>
#include <math.h>

// ---------------------------------------------------------------------------
// FNO forward for MI455X (gfx1250, wave32).
// Heavy stages on the WMMA pipe (v_wmma_f32_16x16x4_f32), activations staged
// through LDS with CDNA5 async global->LDS copies (ASYNCcnt-tracked):
//   - truncated forward DFT  (only the 24x12 corner modes are needed)
//   - sparse inverse DFT reconstruction
//   - 1x1 convs fused with spectral add + bias + erf-GELU
//   - fc1 (32->128) fused with GELU and the fc2 (128->1) reduction
// Workspace: h[64MB] + spec[64MB] + xft/oft[~1.2MB] + basis tables (~160KB).
// ---------------------------------------------------------------------------

#define B_    8
#define C_    32
#define HW    65536
#define NMODE 288   // 24 kx-rows * 12 ky-cols

typedef float v2f __attribute__((ext_vector_type(2)));
typedef float v8f __attribute__((ext_vector_type(8)));

__device__ __forceinline__ float gelu_erf(float x) {
  return 0.5f * x * (1.0f + erff(x * 0.70710678118654752f));
}

#define WMMA_F32(A, Bm, Cm) \
  __builtin_amdgcn_wmma_f32_16x16x4_f32(false, (A), false, (Bm), (short)0, (Cm), false, false)

// CDNA5 async copy global -> LDS (16B per lane), tracked by ASYNCcnt.
// LDS operand is the low 32 bits of the generic shared address (flat LDS
// aperture truncates to addr[31:0], ISA 10.2).
__device__ __forceinline__ void async_g2l_b128(const void* g, const void* l) {
  unsigned long long ga = (unsigned long long)g;
  unsigned la = (unsigned)(unsigned long long)l;
  asm volatile("global_load_async_to_lds_b128 %0, %1, off" :: "v"(la), "v"(ga) : "memory");
}
__device__ __forceinline__ void async_wait0() {
  asm volatile("s_wait_asynccnt 0x0" ::: "memory");
}

// ------------------------- Fourier basis tables ----------------------------
// fxt[w][ky16]  : e^{-2pi i ky w/256}            (fwd along W, zero ky>=12)
// fy [j][h]     : e^{-2pi i kx_j h/256}          (fwd along H, kx_j corner set)
// gy [h][j]     : (1/256) e^{+2pi i kx_j h/256}  (inv along H)
// gx [ky16][w]  : (c_ky/256) e^{+2pi i ky w/256} (inv along W; imag pre-negated)
__global__ void k_basis(float* fxtr, float* fxti, float* fyr, float* fyi,
                        float* gyr, float* gyi, float* gxr, float* gxin) {
  const float STEP = 6.28318530717958647692f / 256.0f;
  int tid = threadIdx.x;
  for (int i = tid; i < 256 * 16; i += 256) {
    int w = i >> 4, ky = i & 15;
    float r = 0.f, im = 0.f;
    if (ky < 12) {
      float a = ((ky * w) & 255) * STEP;
      r = cosf(a); im = -sinf(a);
    }
    fxtr[i] = r; fxti[i] = im;
  }
  for (int i = tid; i < 24 * 256; i += 256) {
    int j = i >> 8, h = i & 255;
    int kx = (j < 12) ? j : (j + 232);        // kx = 244..255 for j = 12..23
    float a = ((kx * h) & 255) * STEP;
    fyr[i] = cosf(a); fyi[i] = -sinf(a);
  }
  for (int i = tid; i < 256 * 24; i += 256) {
    int h = i / 24, j = i % 24;
    int kx = (j < 12) ? j : (j + 232);
    float a = ((kx * h) & 255) * STEP;
    gyr[i] = cosf(a) * (1.0f / 256.0f);
    gyi[i] = sinf(a) * (1.0f / 256.0f);
  }
  for (int i = tid; i < 16 * 256; i += 256) {
    int ky = i >> 8, w = i & 255;
    float r = 0.f, im = 0.f;
    if (ky < 12) {
      float c = (ky == 0 ? 1.0f : 2.0f) * (1.0f / 256.0f);  // Hermitian doubling
      float a = ((ky * w) & 255) * STEP;
      r = c * cosf(a); im = -c * sinf(a);     // pre-negated imag for fused WMMA
    }
    gxr[i] = r; gxin[i] = im;
  }
}

// ----------------------------- fc0 lift (K=4) ------------------------------
__global__ void k_fc0(const float* __restrict__ x, const float* __restrict__ w,
                      const float* __restrict__ bias, float* __restrict__ h) {
  __shared__ float sw[128];
  __shared__ float sb[32];
  int tid = threadIdx.x;
  if (tid < 128) sw[tid] = w[tid];
  if (tid < 32)  sb[tid] = bias[tid];
  __syncthreads();
  int p = blockIdx.x * 256 + tid;          // 0 .. 524287
  int b = p >> 16, pl = p & 65535;
  const float4 v = reinterpret_cast<const float4*>(x)[p];
  float* hb = h + (size_t)b * C_ * HW + pl;
#pragma unroll
  for (int c = 0; c < 32; ++c) {
    hb[(size_t)c * HW] =
        sb[c] + v.x * sw[c] + v.y * sw[32 + c] + v.z * sw[64 + c] + v.w * sw[96 + c];
  }
}

// ---------------- forward truncated DFT: one block per (b,c) ---------------
// Phase 1 (WMMA): T[256][16] = img(256x256) * Fx^T(256x16)  (real, imag)
//   image A-tiles staged to LDS in 256x16 chunks via async global->LDS.
// Phase 2 (VALU): xft[24][12] = Fy(24x256) * T  (complex x complex)
__global__ void k_dft_fwd(const float* __restrict__ hbuf, float* __restrict__ xft,
                          const float* __restrict__ fxtr, const float* __restrict__ fxti,
                          const float* __restrict__ fyr, const float* __restrict__ fyi) {
  __shared__ float Tr[256 * 16];
  __shared__ float Ti[256 * 16];
  __shared__ float sA[256 * 20];   // 256 rows x 16 cols, pitch 20 (bank-spread)
  int tid = threadIdx.x;
  int wv = tid >> 5, lane = tid & 31;
  int lrow = lane & 15, lhi = lane >> 4;
  const float* img = hbuf + (size_t)blockIdx.x * HW;
  int mtb = wv * 2;                // this wave's two 16-row tiles

  v8f accr[2] = {}; v8f acci[2] = {};
  for (int kc = 0; kc < 256; kc += 16) {
    // stage cols kc..kc+15 for all 256 rows (coalesced async b128)
#pragma unroll
    for (int i = 0; i < 4; ++i) {
      int f = i * 256 + tid;               // 0..1023 float4 transfers
      int row = f >> 2, c4 = (f & 3) << 2;
      async_g2l_b128(img + row * 256 + kc + c4, &sA[row * 20 + c4]);
    }
    async_wait0();
    __syncthreads();
#pragma unroll
    for (int k0 = 0; k0 < 16; k0 += 4) {
      int ka = k0 + 2 * lhi;
      int kg = kc + ka;
      v2f br; br[0] = fxtr[kg * 16 + lrow]; br[1] = fxtr[(kg + 1) * 16 + lrow];
      v2f bi; bi[0] = fxti[kg * 16 + lrow]; bi[1] = fxti[(kg + 1) * 16 + lrow];
#pragma unroll
      for (int m = 0; m < 2; ++m) {
        const float* ar = &sA[((mtb + m) * 16 + lrow) * 20 + ka];
        v2f a; a[0] = ar[0]; a[1] = ar[1];
        accr[m] = WMMA_F32(a, br, accr[m]);
        acci[m] = WMMA_F32(a, bi, acci[m]);
      }
    }
    __syncthreads();                        // before next chunk overwrites sA
  }
#pragma unroll
  for (int m = 0; m < 2; ++m)
#pragma unroll
    for (int r = 0; r < 8; ++r) {
      int hh = (mtb + m) * 16 + r + 8 * lhi;
      Tr[hh * 16 + lrow] = accr[m][r];
      Ti[hh * 16 + lrow] = acci[m][r];
    }
  __syncthreads();
  for (int idx = tid; idx < NMODE; idx += 256) {
    int j = idx / 12, ky = idx % 12;
    const float* yr = fyr + j * 256;
    const float* yi = fyi + j * 256;
    float xr = 0.f, xi = 0.f;
    for (int hh = 0; hh < 256; ++hh) {
      float tr = Tr[hh * 16 + ky], ti = Ti[hh * 16 + ky];
      float cr = yr[hh], ci = yi[hh];
      xr += cr * tr - ci * ti;
      xi += cr * ti + ci * tr;
    }
    float* o = xft + ((size_t)blockIdx.x * NMODE + idx) * 2;
    o[0] = xr; o[1] = xi;
  }
}

// ------------- per-mode complex channel mixing (288 tiny GEMMs) ------------
__global__ void k_mix(const float* __restrict__ xft, float* __restrict__ oft,
                      const float* __restrict__ w1, const float* __restrict__ w2) {
  int mode = blockIdx.x;                // j*12 + ky
  int j = mode / 12, ky = mode % 12;
  int tid = threadIdx.x;
  int b = tid >> 5, o = tid & 31;
  const float* wb = (j < 12) ? w1 : w2; // [2][i][o][12][12]
  int jx = (j < 12) ? j : j - 12;
  size_t woff = (size_t)o * 144 + jx * 12 + ky;
  float outr = 0.f, outi = 0.f;
  for (int i = 0; i < 32; ++i) {
    const float* xp = xft + (((size_t)b * 32 + i) * NMODE + mode) * 2;
    float ar = xp[0], ai = xp[1];
    float wr = wb[(size_t)i * 4608 + woff];
    float wi = wb[147456 + (size_t)i * 4608 + woff];
    outr += ar * wr - ai * wi;
    outi += ar * wi + ai * wr;
  }
  float* op = oft + (((size_t)b * 32 + o) * NMODE + mode) * 2;
  op[0] = outr; op[1] = outi;
}

// -------------- sparse inverse DFT: one block per (b,o) image --------------
// Phase 1 (VALU): U[256][16] = Gy(256x24) * oft(24x12)  (complex)
// Phase 2 (WMMA): y = Ur*Gxr + Ui*Gxin   (Re of complex product; imag pre-neg)
__global__ void k_dft_inv(const float* __restrict__ oft, float* __restrict__ spec,
                          const float* __restrict__ gyr, const float* __restrict__ gyi,
                          const float* __restrict__ gxr, const float* __restrict__ gxin) {
  __shared__ float Ur[256 * 16];
  __shared__ float Ui[256 * 16];
  __shared__ float ofr[NMODE];
  __shared__ float ofi[NMODE];
  int tid = threadIdx.x;
  for (int i = tid; i < NMODE; i += 256) {
    const float* p = oft + ((size_t)blockIdx.x * NMODE + i) * 2;
    ofr[i] = p[0]; ofi[i] = p[1];
  }
  __syncthreads();
  {
    int hh = tid;
    const float* gr = gyr + hh * 24;
    const float* gi = gyi + hh * 24;
    for (int ky = 0; ky < 12; ++ky) {
      float ur = 0.f, ui = 0.f;
#pragma unroll
      for (int j = 0; j < 24; ++j) {
        float cr = gr[j], ci = gi[j];
        float vr = ofr[j * 12 + ky], vi = ofi[j * 12 + ky];
        ur += cr * vr - ci * vi;
        ui += cr * vi + ci * vr;
      }
      Ur[hh * 16 + ky] = ur;
      Ui[hh * 16 + ky] = ui;
    }
#pragma unroll
    for (int ky = 12; ky < 16; ++ky) { Ur[hh * 16 + ky] = 0.f; Ui[hh * 16 + ky] = 0.f; }
  }
  __syncthreads();
  int wv = tid >> 5, lane = tid & 31;
  int lrow = lane & 15, lhi = lane >> 4;
  float* out = spec + (size_t)blockIdx.x * HW;
  for (int t = wv; t < 256; t += 8) {
    int mt = t >> 4, nt = t & 15;
    v8f acc = {};
    int arow = (mt * 16 + lrow) * 16;
#pragma unroll
    for (int k0 = 0; k0 < 16; k0 += 4) {
      int ka = k0 + 2 * lhi;
      v2f ar; ar[0] = Ur[arow + ka];  ar[1] = Ur[arow + ka + 1];
      v2f ai; ai[0] = Ui[arow + ka];  ai[1] = Ui[arow + ka + 1];
      v2f br; br[0] = gxr[ka * 256 + nt * 16 + lrow];
              br[1] = gxr[(ka + 1) * 256 + nt * 16 + lrow];
      v2f bi; bi[0] = gxin[ka * 256 + nt * 16 + lrow];
              bi[1] = gxin[(ka + 1) * 256 + nt * 16 + lrow];
      acc = WMMA_F32(ar, br, acc);
      acc = WMMA_F32(ai, bi, acc);
    }
#pragma unroll
    for (int r = 0; r < 8; ++r) {
      int hh = mt * 16 + r + 8 * lhi;
      out[hh * 256 + nt * 16 + lrow] = acc[r];
    }
  }
}

// ------ 1x1 conv (32x32 GEMM over pixels) + spectral add + bias + GELU -----
// Activation tile (32ch x 256px, 32KB) staged to LDS with async copies; all
// reads of hsrc complete before the epilogue writes, so in-place is safe.
__global__ void k_conv_gelu(const float* hsrc, const float* __restrict__ spec,
                            const float* __restrict__ wmat, const float* __restrict__ bias,
                            float* hdst) {
  __shared__ float sB[32 * 256];
  int tid = threadIdx.x;
  int wv = tid >> 5, lane = tid & 31;
  int lrow = lane & 15, lhi = lane >> 4;
  int pg = blockIdx.x * 256;
  int b = pg >> 16;
  int ptile = pg & 65535;
  const float* hb = hsrc + (size_t)b * C_ * HW;

#pragma unroll
  for (int i = 0; i < 8; ++i) {
    int f = i * 256 + tid;                 // 2048 float4 transfers
    int ch = f >> 6, px4 = (f & 63) << 2;
    async_g2l_b128(hb + (size_t)ch * HW + ptile + px4, &sB[ch * 256 + px4]);
  }
  async_wait0();
  __syncthreads();

  int pw = wv * 32;                        // wave's pixel offset in tile
  v8f acc[2][2] = {};
#pragma unroll
  for (int k0 = 0; k0 < 32; k0 += 4) {
    int ka = k0 + 2 * lhi;
    v2f a0, a1;
    a0[0] = wmat[lrow * 32 + ka];        a0[1] = wmat[lrow * 32 + ka + 1];
    a1[0] = wmat[(16 + lrow) * 32 + ka]; a1[1] = wmat[(16 + lrow) * 32 + ka + 1];
    v2f b0, b1;
    b0[0] = sB[ka * 256 + pw + lrow];      b0[1] = sB[(ka + 1) * 256 + pw + lrow];
    b1[0] = sB[ka * 256 + pw + 16 + lrow]; b1[1] = sB[(ka + 1) * 256 + pw + 16 + lrow];
    acc[0][0] = WMMA_F32(a0, b0, acc[0][0]);
    acc[0][1] = WMMA_F32(a0, b1, acc[0][1]);
    acc[1][0] = WMMA_F32(a1, b0, acc[1][0]);
    acc[1][1] = WMMA_F32(a1, b1, acc[1][1]);
  }
  const float* sb = spec + (size_t)b * C_ * HW;
  float* db = hdst + (size_t)b * C_ * HW;
#pragma unroll
  for (int mt = 0; mt < 2; ++mt)
#pragma unroll
    for (int nt = 0; nt < 2; ++nt)
#pragma unroll
      for (int r = 0; r < 8; ++r) {
        int o = mt * 16 + r + 8 * lhi;
        int p = ptile + pw + nt * 16 + lrow;
        size_t off = (size_t)o * HW + p;
        db[off] = gelu_erf(acc[mt][nt][r] + sb[off] + bias[o]);
      }
}

// -------- fused head: fc1 (128x32 WMMA) + GELU + fc2 dot, 16 px/wave -------
__global__ void k_head(const float* __restrict__ hsrc,
                       const float* __restrict__ w1, const float* __restrict__ b1,
                       const float* __restrict__ w2, const float* __restrict__ b2,
                       float* __restrict__ out) {
  __shared__ float sB[32 * 128];           // 32ch x 128px tile (16KB)
  int tid = threadIdx.x;
  int wv = tid >> 5, lane = tid & 31;
  int lrow = lane & 15, lhi = lane >> 4;
  int pg = blockIdx.x * 128;
  int b = pg >> 16;
  int ptile = pg & 65535;
  const float* hb = hsrc + (size_t)b * C_ * HW;

#pragma unroll
  for (int i = 0; i < 4; ++i) {
    int f = i * 256 + tid;                 // 1024 float4 transfers
    int ch = f >> 5, px4 = (f & 31) << 2;
    async_g2l_b128(hb + (size_t)ch * HW + ptile + px4, &sB[ch * 128 + px4]);
  }
  async_wait0();
  __syncthreads();

  int pw = wv * 16;                        // wave's 16 pixels in tile
  v8f acc[8] = {};
#pragma unroll
  for (int k0 = 0; k0 < 32; k0 += 4) {
    int ka = k0 + 2 * lhi;
    v2f bf;
    bf[0] = sB[ka * 128 + pw + lrow];
    bf[1] = sB[(ka + 1) * 128 + pw + lrow];
#pragma unroll
    for (int mt = 0; mt < 8; ++mt) {
      v2f a;  // A[d,c] = fc1_w[c,d], fc1_w row-major [32][128]
      a[0] = w1[(size_t)ka * 128 + mt * 16 + lrow];
      a[1] = w1[(size_t)(ka + 1) * 128 + mt * 16 + lrow];
      acc[mt] = WMMA_F32(a, bf, acc[mt]);
    }
  }
  float partial = 0.f;
#pragma unroll
  for (int mt = 0; mt < 8; ++mt)
#pragma unroll
    for (int r = 0; r < 8; ++r) {
      int d = mt * 16 + r + 8 * lhi;
      partial += gelu_erf(acc[mt][r] + b1[d]) * w2[d];
    }
  partial += __shfl_xor(partial, 16, 32);  // sum lane-halves (wave32)
  if (lhi == 0) out[(size_t)b * HW + ptile + pw + lrow] = partial + b2[0];
}

// ---------------------------------------------------------------------------
extern "C" void kernel_launch(void* const* d_in, const int* in_sizes, int n_in,
                              void* d_out, int out_size, void* d_ws, size_t ws_size,
                              hipStream_t stream) {
  (void)in_sizes; (void)n_in; (void)out_size; (void)ws_size;
  const float* x      = (const float*)d_in[0];
  const float* fc0_w  = (const float*)d_in[1];
  const float* fc0_b  = (const float*)d_in[2];
  const float* sc0_w1 = (const float*)d_in[3];
  const float* sc0_w2 = (const float*)d_in[4];
  const float* sc1_w1 = (const float*)d_in[5];
  const float* sc1_w2 = (const float*)d_in[6];
  const float* w0_w   = (const float*)d_in[7];
  const float* w0_b   = (const float*)d_in[8];
  const float* w1_w   = (const float*)d_in[9];
  const float* w1_b   = (const float*)d_in[10];
  const float* fc1_w  = (const float*)d_in[11];
  const float* fc1_b  = (const float*)d_in[12];
  const float* fc2_w  = (const float*)d_in[13];
  const float* fc2_b  = (const float*)d_in[14];
  float* out = (float*)d_out;

  char* ws = (char*)d_ws;
  size_t off = 0;
  auto alloc = [&](size_t bytes) -> float* {
    float* p = (float*)(ws + off);
    off += (bytes + 255) & ~(size_t)255;
    return p;
  };
  float* hbuf = alloc((size_t)B_ * C_ * HW * 4);   // 64 MB activations
  float* spec = alloc((size_t)B_ * C_ * HW * 4);   // 64 MB spectral branch
  float* xft  = alloc((size_t)B_ * C_ * NMODE * 2 * 4);
  float* oft  = alloc((size_t)B_ * C_ * NMODE * 2 * 4);
  float* fxtr = alloc(256 * 16 * 4); float* fxti = alloc(256 * 16 * 4);
  float* fyr  = alloc(24 * 256 * 4); float* fyi  = alloc(24 * 256 * 4);
  float* gyr  = alloc(256 * 24 * 4); float* gyi  = alloc(256 * 24 * 4);
  float* gxr  = alloc(16 * 256 * 4); float* gxin = alloc(16 * 256 * 4);

  dim3 blk(256);
  k_basis<<<1, blk, 0, stream>>>(fxtr, fxti, fyr, fyi, gyr, gyi, gxr, gxin);
  k_fc0<<<2048, blk, 0, stream>>>(x, fc0_w, fc0_b, hbuf);
  // layer 0
  k_dft_fwd<<<256, blk, 0, stream>>>(hbuf, xft, fxtr, fxti, fyr, fyi);
  k_mix<<<288, blk, 0, stream>>>(xft, oft, sc0_w1, sc0_w2);
  k_dft_inv<<<256, blk, 0, stream>>>(oft, spec, gyr, gyi, gxr, gxin);
  k_conv_gelu<<<2048, blk, 0, stream>>>(hbuf, spec, w0_w, w0_b, hbuf);
  // layer 1
  k_dft_fwd<<<256, blk, 0, stream>>>(hbuf, xft, fxtr, fxti, fyr, fyi);
  k_mix<<<288, blk, 0, stream>>>(xft, oft, sc1_w1, sc1_w2);
  k_dft_inv<<<256, blk, 0, stream>>>(oft, spec, gyr, gyi, gxr, gxin);
  k_conv_gelu<<<2048, blk, 0, stream>>>(hbuf, spec, w1_w, w1_b, hbuf);
  // head
  k_head<<<4096, blk, 0, stream>>>(hbuf, fc1_w, fc1_b, fc2_w, fc2_b, out);
}